// RelGraphConvHetero_86406152061455
// MI455X (gfx1250) — compile-verified
//
#include <hip/hip_runtime.h>
#include <hip/hip_bf16.h>

// Problem constants (match reference)
#define NN   50000
#define INF  128
#define OUTF 128
#define RR   8
#define BB   4

typedef float v2f __attribute__((ext_vector_type(2)));
typedef float v8f __attribute__((ext_vector_type(8)));

// ---------------------------------------------------------------------------
// ws[r,i,o] = sum_b w_comp[r,b] * weight[b,i,o]      (R*IN*OUT = 131072 elems)
// ---------------------------------------------------------------------------
__global__ void rgcn_ws_kernel(const float* __restrict__ w_comp,
                               const float* __restrict__ weight,
                               float* __restrict__ ws) {
    int i  = blockIdx.x * 256 + threadIdx.x;       // < R*IN*OUT
    int r  = i >> 14;                              // IN*OUT = 16384
    int io = i & 16383;
    float s = 0.f;
#pragma unroll
    for (int b = 0; b < BB; ++b)
        s += w_comp[r * BB + b] * weight[b * (INF * OUTF) + io];
    ws[i] = s;
}

// ---------------------------------------------------------------------------
// h = x @ ws_r  via V_WMMA_F32_16X16X4_F32.
// One wave = one 16x16 C tile. Block = 256 threads = 8 waves = 8 col tiles
// covering one 16-row stripe of x. Grid = N/16 = 3125 blocks.
// A 16x4 layout: lane L holds row L%16, K = 2*(L/16)+{0,1}  (float2)
// B 4x16 layout: lane L holds col L%16, K = 2*(L/16)+{0,1}
// C/D: component v holds row v + 8*(L/16), col L%16.
// ---------------------------------------------------------------------------
__global__ void rgcn_gemm_kernel(const float* __restrict__ x,
                                 const float* __restrict__ wsr,
                                 float* __restrict__ h) {
    const int lane  = threadIdx.x & 31;
    const int wave  = threadIdx.x >> 5;
    const int mbase = blockIdx.x * 16;
    const int nbase = wave * 16;
    const int lr    = lane & 15;          // row/col within tile
    const int khalf = (lane >> 4) * 2;    // 0 or 2

    const float* arow = x + (mbase + lr) * INF;
    const float* bcol = wsr + (nbase + lr);

    v8f c = {0.f, 0.f, 0.f, 0.f, 0.f, 0.f, 0.f, 0.f};
#pragma unroll
    for (int kk = 0; kk < INF; kk += 4) {
        v2f a = *(const v2f*)(arow + kk + khalf);               // contiguous pair
        v2f b;
        b.x = bcol[(kk + khalf) * OUTF];
        b.y = bcol[(kk + khalf + 1) * OUTF];
        c = __builtin_amdgcn_wmma_f32_16x16x4_f32(
                /*neg_a=*/false, a, /*neg_b=*/false, b,
                /*c_mod=*/(short)0, c, /*reuse_a=*/false, /*reuse_b=*/false);
    }

    float* hrow = h + nbase + lr;
#pragma unroll
    for (int v = 0; v < 8; ++v) {
        int m = mbase + v + 8 * (lane >> 4);
        hrow[m * OUTF] = c[v];
    }
}

// ---------------------------------------------------------------------------
// Edge scatter: one wave per edge. Gather 512B row of h[src], atomically add
// into acc[dst] (L2-resident: acc is 25.6MB << 192MB L2). Lane0 counts degree.
// ---------------------------------------------------------------------------
__global__ void rgcn_scatter_kernel(const float* __restrict__ h,
                                    const int* __restrict__ src,
                                    const int* __restrict__ dst,
                                    float* __restrict__ acc,
                                    float* __restrict__ deg,
                                    int nedges) {
    const int lane = threadIdx.x & 31;
    const int wave = threadIdx.x >> 5;
    const int e    = blockIdx.x * 8 + wave;
    if (e >= nedges) return;
    const int s = src[e];
    const int d = dst[e];
    const float4 m = ((const float4*)(h + (long)s * OUTF))[lane];
    float* a = acc + (long)d * OUTF + lane * 4;
    atomicAdd(a + 0, m.x);
    atomicAdd(a + 1, m.y);
    atomicAdd(a + 2, m.z);
    atomicAdd(a + 3, m.w);
    if (lane == 0) atomicAdd(deg + d, 1.0f);
}

// ---------------------------------------------------------------------------
// out += acc / max(deg, 1)
// ---------------------------------------------------------------------------
__global__ void rgcn_accum_kernel(const float* __restrict__ acc,
                                  const float* __restrict__ deg,
                                  float* __restrict__ out) {
    int i = blockIdx.x * 256 + threadIdx.x;        // < N*OUT
    float dg = deg[i >> 7];                        // OUT = 128
    out[i] += acc[i] / fmaxf(dg, 1.0f);
}

// ---------------------------------------------------------------------------
// out = relu(out + bias)
// ---------------------------------------------------------------------------
__global__ void rgcn_bias_relu_kernel(float* __restrict__ out,
                                      const float* __restrict__ bias) {
    int i = blockIdx.x * 256 + threadIdx.x;        // < N*OUT
    out[i] = fmaxf(out[i] + bias[i & (OUTF - 1)], 0.0f);
}

// ---------------------------------------------------------------------------
extern "C" void kernel_launch(void* const* d_in, const int* in_sizes, int n_in,
                              void* d_out, int out_size, void* d_ws, size_t ws_size,
                              hipStream_t stream) {
    const float* x      = (const float*)d_in[0];   // [N, IN]
    const float* weight = (const float*)d_in[1];   // [B, IN, OUT]
    const float* w_comp = (const float*)d_in[2];   // [R, B]
    const float* h_bias = (const float*)d_in[3];   // [OUT]
    const int*   src    = (const int*)d_in[4];     // [R, E]
    const int*   dst    = (const int*)d_in[5];     // [R, E]
    float*       out    = (float*)d_out;           // [N, OUT]

    const int E = in_sizes[4] / RR;

    // Scratch layout inside d_ws (all f32), 256B-aligned regions:
    //   ws_all : R*IN*OUT            =    512 KB
    //   h      : N*OUT               =  25.6 MB
    //   acc    : N*OUT               =  25.6 MB
    //   deg    : N                   =   0.2 MB
    char* base = (char*)d_ws;
    size_t off = 0;
    float* ws_all = (float*)(base + off); off += (size_t)RR * INF * OUTF * 4;
    off = (off + 255) & ~(size_t)255;
    float* h      = (float*)(base + off); off += (size_t)NN * OUTF * 4;
    off = (off + 255) & ~(size_t)255;
    float* acc    = (float*)(base + off); off += (size_t)NN * OUTF * 4;
    off = (off + 255) & ~(size_t)255;
    float* deg    = (float*)(base + off);

    // out accumulates across relations -> zero it first
    hipMemsetAsync(out, 0, (size_t)NN * OUTF * 4, stream);

    // per-relation weights
    rgcn_ws_kernel<<<(RR * INF * OUTF) / 256, 256, 0, stream>>>(w_comp, weight, ws_all);

    const int gemm_blocks    = NN / 16;                 // 3125
    const int scatter_blocks = (E + 7) / 8;
    const int elem_blocks    = (NN * OUTF) / 256;       // 25000

    for (int r = 0; r < RR; ++r) {
        hipMemsetAsync(acc, 0, (size_t)NN * OUTF * 4, stream);
        hipMemsetAsync(deg, 0, (size_t)NN * 4, stream);

        rgcn_gemm_kernel<<<gemm_blocks, 256, 0, stream>>>(
            x, ws_all + (size_t)r * INF * OUTF, h);

        rgcn_scatter_kernel<<<scatter_blocks, 256, 0, stream>>>(
            h, src + (size_t)r * E, dst + (size_t)r * E, acc, deg, E);

        rgcn_accum_kernel<<<elem_blocks, 256, 0, stream>>>(acc, deg, out);
    }

    rgcn_bias_relu_kernel<<<elem_blocks, 256, 0, stream>>>(out, h_bias);
}